// AnalyticalLinearSystem_26027501814665
// MI455X (gfx1250) — compile-verified
//
#include <hip/hip_runtime.h>

typedef float v2f __attribute__((ext_vector_type(2)));
typedef float v8f __attribute__((ext_vector_type(8)));

#define DIM     256
#define BATCH   4096
#define TSTEPS  100
#define DT      0.05f
#define MATELEMS (DIM * DIM)

// traj tiling: each block = 8 waves, 128 batch rows x 64 k columns, E-tile in LDS
#define BT 128
#define KT 64

// ---------------------------------------------------------------------------
// Fragment loaders for V_WMMA_F32_16X16X4_F32 (wave32).
//
// A 16x4 frag (ISA 7.12.2, 32-bit A): lane L holds row (L&15); VGPR0/1 hold
// K = 2*(L>>4) and 2*(L>>4)+1  -> one aligned 8-byte load per lane.
// Same pattern serves the B-frag of an N*T GEMM (B[kk,n] = E[k0+n, d0+kk]).
__device__ __forceinline__ v2f frag_pair(const float* __restrict__ base,
                                         int row0, int col0) {
  const int lane = threadIdx.x & 31;
  const int r = row0 + (lane & 15);
  const int c = col0 + ((lane >> 4) << 1);
  return *reinterpret_cast<const v2f*>(base + r * DIM + c);
}

// B 4x16 frag for an N*N GEMM: B[kk,n] = T[d0+kk, j0+n]; rows strided -> 2 b32.
__device__ __forceinline__ v2f frag_b_nn(const float* __restrict__ base,
                                         int k0, int n0) {
  const int lane = threadIdx.x & 31;
  const int n = n0 + (lane & 15);
  const int k = k0 + ((lane >> 4) << 1);
  v2f v;
  v.x = base[k * DIM + n];
  v.y = base[(k + 1) * DIM + n];
  return v;
}

__device__ __forceinline__ v8f wmma_f32(v2f a, v2f b, v8f c) {
  // 8 args: (neg_a, A, neg_b, B, c_mod, C, reuse_a, reuse_b)
  return __builtin_amdgcn_wmma_f32_16x16x4_f32(false, a, false, b,
                                               (short)0, c, false, false);
}

// gfx1250 async global->LDS copy, 16B per lane, tracked by ASYNCcnt.
// VDST = per-lane LDS byte address, VADDR = per-lane 64-bit global address.
__device__ __forceinline__ void async_copy_b128(unsigned lds_addr,
                                                const float* gaddr) {
  asm volatile("global_load_async_to_lds_b128 %0, %1, off"
               :: "v"(lds_addr), "v"(gaddr)
               : "memory");
}

__device__ __forceinline__ void wait_asynccnt0() {
  asm volatile("s_wait_asynccnt 0x0" ::: "memory");
}

// ---------------------------------------------------------------------------
// Step 0: Bm = dt*A ; T = I + (dt/8)*A   (innermost Horner term, N = 8)
__global__ void prep_kernel(const float* __restrict__ A,
                            float* __restrict__ Bm,
                            float* __restrict__ T) {
  const int idx = blockIdx.x * blockDim.x + threadIdx.x;   // 65536 threads
  const int row = idx >> 8, col = idx & 255;
  const float a = A[idx];
  Bm[idx] = DT * a;
  T[idx] = ((row == col) ? 1.0f : 0.0f) + (DT / 8.0f) * a;
}

// Horner step: Tout = I + invk * (Bm @ Tin).  One wave per 16x16 tile.
__global__ void expm_step_kernel(const float* __restrict__ Bm,
                                 const float* __restrict__ Tin,
                                 float* __restrict__ Tout, float invk) {
  const int i0 = blockIdx.y << 4;
  const int j0 = blockIdx.x << 4;
  v8f acc = {};
  #pragma unroll 8
  for (int d0 = 0; d0 < DIM; d0 += 4) {
    v2f a = frag_pair(Bm, i0, d0);
    v2f b = frag_b_nn(Tin, d0, j0);
    acc = wmma_f32(a, b, acc);
  }
  const int lane  = threadIdx.x & 31;
  const int n     = j0 + (lane & 15);
  const int mbase = i0 + ((lane >> 4) << 3);
  #pragma unroll
  for (int i = 0; i < 8; ++i) {
    const int m = mbase + i;
    Tout[m * DIM + n] = invk * acc[i] + ((m == n) ? 1.0f : 0.0f);
  }
}

// E_0 = I ; E_1 = M
__global__ void initE_kernel(const float* __restrict__ M,
                             float* __restrict__ E) {
  const int idx = blockIdx.x * blockDim.x + threadIdx.x;
  const int row = idx >> 8, col = idx & 255;
  E[idx] = (row == col) ? 1.0f : 0.0f;
  E[MATELEMS + idx] = M[idx];
}

// Doubling stage: for j = 1..gridDim.z, E_{p+j} = E_j @ E_p (batched GEMM).
// Reads indices <= p, writes indices > p -> no overlap within a launch.
__global__ void power_stage_kernel(float* __restrict__ E, int p) {
  const int j = blockIdx.z + 1;
  const float* Ej = E + (size_t)j * MATELEMS;
  const float* Ep = E + (size_t)p * MATELEMS;
  float*       Eo = E + (size_t)(p + j) * MATELEMS;
  const int i0 = blockIdx.y << 4;
  const int j0 = blockIdx.x << 4;
  v8f acc = {};
  #pragma unroll 8
  for (int d0 = 0; d0 < DIM; d0 += 4) {
    v2f a = frag_pair(Ej, i0, d0);
    v2f b = frag_b_nn(Ep, d0, j0);
    acc = wmma_f32(a, b, acc);
  }
  const int lane  = threadIdx.x & 31;
  const int n     = j0 + (lane & 15);
  const int mbase = i0 + ((lane >> 4) << 3);
  #pragma unroll
  for (int i = 0; i < 8; ++i)
    Eo[(mbase + i) * DIM + n] = acc[i];
}

// Final: out[b,t,k] = sum_d X0[b,d] * E_t[k,d]  (C = X0 @ E_t^T).
// Block = 256 threads (8 waves): 128 b-rows x 64 k-cols. The 64x256 E-tile is
// staged once into LDS with async b128 copies and shared by all 8 waves; each
// wave computes a 16x64 tile with an A-fragment reused across 4 accumulators.
__global__ void __launch_bounds__(256)
traj_kernel(const float* __restrict__ X0,
            const float* __restrict__ E,
            float* __restrict__ out) {
  __shared__ __align__(16) float Esh[KT * DIM];   // 64 KB

  const int tid  = threadIdx.x;
  const int wave = tid >> 5;
  const int lane = tid & 31;
  const int b0 = blockIdx.x * BT + wave * 16;
  const int k0 = blockIdx.y * KT;
  const int t  = blockIdx.z;
  // rows k0..k0+63 of E_t: one contiguous 64KB span
  const float* Et = E + (size_t)t * MATELEMS + (size_t)k0 * DIM;

  // cooperative async copy: 256 threads * 16 iters * 16B = 64KB
  {
    const unsigned lds_base = (unsigned)(size_t)(void*)Esh;  // addr[31:0] = LDS offset
    #pragma unroll
    for (int j = 0; j < (KT * DIM) / (256 * 4); ++j) {       // 16
      const int felt = j * 1024 + tid * 4;                   // float index
      async_copy_b128(lds_base + (unsigned)(felt * 4), Et + felt);
    }
    wait_asynccnt0();   // this wave's async loads landed in LDS
  }
  __syncthreads();      // all waves' tiles visible

  v8f acc0 = {}, acc1 = {}, acc2 = {}, acc3 = {};
  const int kr = lane & 15;
  #pragma unroll 4
  for (int d0 = 0; d0 < DIM; d0 += 4) {
    v2f a = frag_pair(X0, b0, d0);
    const int cc = d0 + ((lane >> 4) << 1);
    v2f e0 = *reinterpret_cast<const v2f*>(&Esh[(kr     ) * DIM + cc]);
    v2f e1 = *reinterpret_cast<const v2f*>(&Esh[(kr + 16) * DIM + cc]);
    v2f e2 = *reinterpret_cast<const v2f*>(&Esh[(kr + 32) * DIM + cc]);
    v2f e3 = *reinterpret_cast<const v2f*>(&Esh[(kr + 48) * DIM + cc]);
    acc0 = wmma_f32(a, e0, acc0);
    acc1 = wmma_f32(a, e1, acc1);
    acc2 = wmma_f32(a, e2, acc2);
    acc3 = wmma_f32(a, e3, acc3);
  }

  const int mbase = b0 + ((lane >> 4) << 3);
  #pragma unroll
  for (int i = 0; i < 8; ++i) {
    const size_t o = (size_t)(mbase + i) * (TSTEPS * DIM)
                   + (size_t)t * DIM + (size_t)(k0 + kr);
    out[o]      = acc0[i];
    out[o + 16] = acc1[i];
    out[o + 32] = acc2[i];
    out[o + 48] = acc3[i];
  }
}

// ---------------------------------------------------------------------------
extern "C" void kernel_launch(void* const* d_in, const int* in_sizes, int n_in,
                              void* d_out, int out_size, void* d_ws, size_t ws_size,
                              hipStream_t stream) {
  (void)in_sizes; (void)n_in; (void)out_size; (void)ws_size;
  const float* X0 = (const float*)d_in[0];   // [4096, 256]
  const float* A  = (const float*)d_in[1];   // [256, 256]
  float* out = (float*)d_out;                // [4096, 100, 256]
  float* ws  = (float*)d_ws;

  float* Bm = ws;                    // dt*A
  float* T0 = ws + MATELEMS;         // Horner ping
  float* T1 = ws + 2 * MATELEMS;     // Horner pong
  float* E  = ws + 3 * MATELEMS;     // E_0..E_99 (100 * 256KB)

  // M = expm(dt*A): 8-term Taylor via Horner (remainder ~1e-16 for ||dtA||~0.07)
  prep_kernel<<<dim3(MATELEMS / 256), 256, 0, stream>>>(A, Bm, T0);
  float* tin = T0; float* tout = T1;
  for (int k = 7; k >= 1; --k) {
    expm_step_kernel<<<dim3(16, 16), 32, 0, stream>>>(Bm, tin, tout,
                                                      1.0f / (float)k);
    float* tmp = tin; tin = tout; tout = tmp;
  }

  // E_0 = I, E_1 = M, then log-depth doubling: 7 stages to reach E_99.
  initE_kernel<<<dim3(MATELEMS / 256), 256, 0, stream>>>(tin, E);
  int n = 1;
  while (n < TSTEPS - 1) {
    const int rem = (TSTEPS - 1) - n;
    const int count = n < rem ? n : rem;
    power_stage_kernel<<<dim3(16, 16, count), 32, 0, stream>>>(E, n);
    n += count;
  }

  // Fully parallel batched GEMM over (b-supertiles, k-groups, t).
  traj_kernel<<<dim3(BATCH / BT, DIM / KT, TSTEPS), 256, 0, stream>>>(X0, E, out);
}